// LatentRelationCapturerGAT_89550068122092
// MI455X (gfx1250) — compile-verified
//
#include <hip/hip_runtime.h>
#include <stdint.h>

typedef __attribute__((ext_vector_type(2))) float v2f;
typedef __attribute__((ext_vector_type(8))) float v8f;

#define NN     50000
#define EE     500000
#define IN_F   256
#define OUT_F  64
#define HH     8
#define ETOT   (EE + NN)
#define NEG_SLOPE 0.2f

// order-preserving float <-> uint map for atomic max on signed floats
__device__ __forceinline__ unsigned f2key(float f) {
  unsigned u = __float_as_uint(f);
  return (u & 0x80000000u) ? ~u : (u | 0x80000000u);
}
__device__ __forceinline__ float key2f(unsigned k) {
  unsigned u = (k & 0x80000000u) ? (k & 0x7FFFFFFFu) : ~k;
  return __uint_as_float(u);
}

__device__ __forceinline__ void edge_decode(const long long* __restrict__ ei,
                                            int e, int& src, int& dst) {
  if (e < EE) {
    src = (int)ei[e];
    dst = (int)ei[(size_t)EE + e];
  } else {          // self loop
    src = dst = e - EE;
  }
}

// ---------------------------------------------------------------------------
// Kernel 1: xp[h] = x @ W[h]  via native FP32 WMMA (V_WMMA_F32_16X16X4_F32)
// Block = 256 threads = 8 waves; each wave owns a 32x64 tile (8 accumulators,
// 2 A-fragments sharing 4 B-fragments -> 8 WMMAs per K-step).
// W[h] staged in LDS pair-interleaved: lWp[k/2][n] = {W[k][n], W[k+1][n]}
// so each B fragment is ONE aligned ds_load_b64 into an even VGPR pair.
// ---------------------------------------------------------------------------
__global__ void __launch_bounds__(256) gat_gemm_xp(
    const float* __restrict__ x, const float* __restrict__ W,
    float* __restrict__ xp) {
  __shared__ __align__(16) v2f lWp[(IN_F / 2) * OUT_F];   // 128*64*8B = 64 KB

  const int head = blockIdx.y;
  const int bm   = blockIdx.x * 256;           // 8 waves * 32 rows
  const float* Wh = W + (size_t)head * IN_F * OUT_F;

  // cooperative pair-interleave fill: 128 kp-rows x 16 groups of 4 cols
  for (int i = threadIdx.x; i < (IN_F / 2) * (OUT_F / 4); i += 256) {
    const int kp = i >> 4;            // 0..127
    const int n0 = (i & 15) << 2;     // 0,4,...,60
    const float4 r0 = *(const float4*)(Wh + (size_t)(2 * kp)     * OUT_F + n0);
    const float4 r1 = *(const float4*)(Wh + (size_t)(2 * kp + 1) * OUT_F + n0);
    v2f* dst = lWp + kp * OUT_F + n0;
    dst[0] = (v2f){r0.x, r1.x};
    dst[1] = (v2f){r0.y, r1.y};
    dst[2] = (v2f){r0.z, r1.z};
    dst[3] = (v2f){r0.w, r1.w};
  }
  __syncthreads();

  const int wave = threadIdx.x >> 5;
  const int lane = threadIdx.x & 31;
  const int half = lane >> 4;                  // lanes 16-31 hold K+2,K+3
  const int l16  = lane & 15;

  // A rows for the two 16-row tiles (clamped: EXEC stays all-1s for WMMA)
  const int arow0_ = bm + wave * 32 + l16;
  const int arow1_ = arow0_ + 16;
  const int arow0  = arow0_ < NN ? arow0_ : NN - 1;
  const int arow1  = arow1_ < NN ? arow1_ : NN - 1;
  const float* xr0 = x + (size_t)arow0 * IN_F;
  const float* xr1 = x + (size_t)arow1 * IN_F;

  v8f t0a = {}, t0b = {}, t0c = {}, t0d = {};
  v8f t1a = {}, t1b = {}, t1c = {}, t1d = {};

  for (int k0 = 0; k0 < IN_F; k0 += 4) {
    const int ka = k0 + half * 2;
    const v2f a0 = *(const v2f*)(xr0 + ka);
    const v2f a1 = *(const v2f*)(xr1 + ka);

    const v2f* wrow = lWp + (k0 / 2 + half) * OUT_F + l16;
    const v2f b0 = wrow[0];
    const v2f b1 = wrow[16];
    const v2f b2 = wrow[32];
    const v2f b3 = wrow[48];

    t0a = __builtin_amdgcn_wmma_f32_16x16x4_f32(false, a0, false, b0, (short)0, t0a, false, false);
    t0b = __builtin_amdgcn_wmma_f32_16x16x4_f32(false, a0, false, b1, (short)0, t0b, false, false);
    t0c = __builtin_amdgcn_wmma_f32_16x16x4_f32(false, a0, false, b2, (short)0, t0c, false, false);
    t0d = __builtin_amdgcn_wmma_f32_16x16x4_f32(false, a0, false, b3, (short)0, t0d, false, false);
    t1a = __builtin_amdgcn_wmma_f32_16x16x4_f32(false, a1, false, b0, (short)0, t1a, false, false);
    t1b = __builtin_amdgcn_wmma_f32_16x16x4_f32(false, a1, false, b1, (short)0, t1b, false, false);
    t1c = __builtin_amdgcn_wmma_f32_16x16x4_f32(false, a1, false, b2, (short)0, t1c, false, false);
    t1d = __builtin_amdgcn_wmma_f32_16x16x4_f32(false, a1, false, b3, (short)0, t1d, false, false);
  }

  // C/D layout: VGPR r -> row (r + half*8), col = l16 (+16 per n-tile)
  float* xph = xp + (size_t)head * NN * OUT_F;
#pragma unroll
  for (int r = 0; r < 8; ++r) {
    const int grow0 = bm + wave * 32 + r + half * 8;
    const int grow1 = grow0 + 16;
    if (grow0 < NN) {
      float* o = xph + (size_t)grow0 * OUT_F + l16;
      o[0]  = t0a[r];
      o[16] = t0b[r];
      o[32] = t0c[r];
      o[48] = t0d[r];
    }
    if (grow1 < NN) {
      float* o = xph + (size_t)grow1 * OUT_F + l16;
      o[0]  = t1a[r];
      o[16] = t1b[r];
      o[32] = t1c[r];
      o[48] = t1d[r];
    }
  }
}

// ---------------------------------------------------------------------------
// Kernel 2: als/ald = xp . a_src / a_dst (wave32 shfl reduction),
//           init m_bits=0, denom=0, out = b (broadcast).
// One wave per (head,node).
// ---------------------------------------------------------------------------
__global__ void __launch_bounds__(256) gat_logits_init(
    const float* __restrict__ xp, const float* __restrict__ a_src,
    const float* __restrict__ a_dst, const float* __restrict__ b,
    float* __restrict__ als, float* __restrict__ ald,
    unsigned* __restrict__ mbits, float* __restrict__ denom,
    float* __restrict__ out) {
  const int wid  = blockIdx.x * 8 + (threadIdx.x >> 5);
  const int lane = threadIdx.x & 31;
  if (wid >= HH * NN) return;
  const int h = wid / NN;

  const float* row = xp + (size_t)wid * OUT_F;
  const float v0 = row[lane];
  const float v1 = row[lane + 32];

  float s = v0 * a_src[h * OUT_F + lane] + v1 * a_src[h * OUT_F + 32 + lane];
  float d = v0 * a_dst[h * OUT_F + lane] + v1 * a_dst[h * OUT_F + 32 + lane];
#pragma unroll
  for (int m = 16; m >= 1; m >>= 1) {
    s += __shfl_xor(s, m, 32);
    d += __shfl_xor(d, m, 32);
  }
  if (lane == 0) {
    als[wid]   = s;
    ald[wid]   = d;
    mbits[wid] = 0u;      // below every real key; self-loop guarantees update
    denom[wid] = 0.0f;
  }
  float* orow = out + (size_t)wid * OUT_F;
  orow[lane]      = b[h * OUT_F + lane];
  orow[lane + 32] = b[h * OUT_F + lane + 32];
}

// ---------------------------------------------------------------------------
// Kernel 3: segment max of leaky-relu logits via atomicMax on ordered keys.
// ---------------------------------------------------------------------------
__global__ void __launch_bounds__(256) gat_edge_max(
    const long long* __restrict__ ei, const float* __restrict__ als,
    const float* __restrict__ ald, unsigned* __restrict__ mbits) {
  const long long tid = (long long)blockIdx.x * 256 + threadIdx.x;
  if (tid >= (long long)ETOT * HH) return;
  const int e = (int)(tid % ETOT);
  const int h = (int)(tid / ETOT);
  int src, dst;
  edge_decode(ei, e, src, dst);
  const float v  = als[h * NN + src] + ald[h * NN + dst];
  const float lr = v > 0.0f ? v : v * NEG_SLOPE;
  atomicMax(mbits + h * NN + dst, f2key(lr));
}

// ---------------------------------------------------------------------------
// Kernel 4: denom[dst] += exp(e - m[dst])  (f32 atomic add in L2)
// ---------------------------------------------------------------------------
__global__ void __launch_bounds__(256) gat_edge_sum(
    const long long* __restrict__ ei, const float* __restrict__ als,
    const float* __restrict__ ald, const unsigned* __restrict__ mbits,
    float* __restrict__ denom) {
  const long long tid = (long long)blockIdx.x * 256 + threadIdx.x;
  if (tid >= (long long)ETOT * HH) return;
  const int e = (int)(tid % ETOT);
  const int h = (int)(tid / ETOT);
  int src, dst;
  edge_decode(ei, e, src, dst);
  const float v  = als[h * NN + src] + ald[h * NN + dst];
  const float lr = v > 0.0f ? v : v * NEG_SLOPE;
  const float m  = key2f(mbits[h * NN + dst]);
  atomicAdd(denom + h * NN + dst, __expf(lr - m));
}

// ---------------------------------------------------------------------------
// Kernel 5: out[h,dst,:] += alpha * xp[h,src,:]   (one wave per edge-head,
// 2 f32 atomic adds per lane). global_prefetch_b8 issued on the xp row to
// overlap the dependent scalar chain (als/ald/m/denom loads).
// ---------------------------------------------------------------------------
__global__ void __launch_bounds__(256) gat_aggregate(
    const long long* __restrict__ ei, const float* __restrict__ als,
    const float* __restrict__ ald, const unsigned* __restrict__ mbits,
    const float* __restrict__ denom, const float* __restrict__ xp,
    float* __restrict__ out) {
  const long long wid = (long long)blockIdx.x * 8 + (threadIdx.x >> 5);
  const int lane = threadIdx.x & 31;
  if (wid >= (long long)ETOT * HH) return;
  const int e = (int)(wid % ETOT);
  const int h = (int)(wid / ETOT);
  int src, dst;
  edge_decode(ei, e, src, dst);

  const float* xr = xp  + ((size_t)h * NN + src) * OUT_F;
  __builtin_prefetch(xr + lane, 0, 1);          // global_prefetch_b8

  const float v     = als[h * NN + src] + ald[h * NN + dst];
  const float lr    = v > 0.0f ? v : v * NEG_SLOPE;
  const float m     = key2f(mbits[h * NN + dst]);
  const float alpha = __expf(lr - m) / denom[h * NN + dst];

  float* o = out + ((size_t)h * NN + dst) * OUT_F;
  atomicAdd(o + lane,      alpha * xr[lane]);
  atomicAdd(o + lane + 32, alpha * xr[lane + 32]);
}

// ---------------------------------------------------------------------------
extern "C" void kernel_launch(void* const* d_in, const int* in_sizes, int n_in,
                              void* d_out, int out_size, void* d_ws, size_t ws_size,
                              hipStream_t stream) {
  const float*     x     = (const float*)d_in[0];
  const long long* ei    = (const long long*)d_in[1];   // int64 edge_index [2,E]
  const float*     W     = (const float*)d_in[2];       // [H, IN, OUT]
  const float*     a_src = (const float*)d_in[3];       // [H, OUT]
  const float*     a_dst = (const float*)d_in[4];       // [H, OUT]
  const float*     b     = (const float*)d_in[5];       // [H, OUT]
  float*           out   = (float*)d_out;               // [H, N, OUT]

  // workspace layout (floats): xp | als | ald | mbits | denom  (~108.8 MB)
  float*    xp    = (float*)d_ws;                               // H*N*OUT
  float*    als   = xp + (size_t)HH * NN * OUT_F;               // H*N
  float*    ald   = als + (size_t)HH * NN;                      // H*N
  unsigned* mbits = (unsigned*)(ald + (size_t)HH * NN);         // H*N
  float*    denom = (float*)(mbits + (size_t)HH * NN);          // H*N

  dim3 g1((NN + 255) / 256, HH);
  gat_gemm_xp<<<g1, 256, 0, stream>>>(x, W, xp);

  gat_logits_init<<<(HH * NN + 7) / 8, 256, 0, stream>>>(
      xp, a_src, a_dst, b, als, ald, mbits, denom, out);

  const long long tot = (long long)ETOT * HH;   // 4.4M edge-heads
  gat_edge_max<<<(unsigned)((tot + 255) / 256), 256, 0, stream>>>(ei, als, ald, mbits);
  gat_edge_sum<<<(unsigned)((tot + 255) / 256), 256, 0, stream>>>(ei, als, ald, mbits, denom);
  gat_aggregate<<<(unsigned)((tot + 7) / 8), 256, 0, stream>>>(
      ei, als, ald, mbits, denom, xp, out);
}